// EmbeddingClassifier_87754771791930
// MI455X (gfx1250) — compile-verified
//
#include <hip/hip_runtime.h>
#include <math.h>

typedef float v2f __attribute__((ext_vector_type(2)));
typedef float v8f __attribute__((ext_vector_type(8)));

#define NB   32      // batch
#define SS   4096    // seq len
#define DD   768     // model dim
#define K2   1536    // 2*D  (GEMM K)
#define G4   3072    // 4*D  per direction
#define NCAT 6144    // both directions concatenated (GEMM N)
#define NLAB 17

// ---------------------------------------------------------------- gather ----
__global__ void gather_pooled_kernel(const float* __restrict__ emb,
                                     const int*   __restrict__ att,
                                     const int*   __restrict__ dom_ids,
                                     const int*   __restrict__ e1,
                                     const int*   __restrict__ e2,
                                     const float* __restrict__ dset,
                                     float*       __restrict__ pooled)
{
    const int b  = blockIdx.x;
    const int p1 = e1[b], p2 = e2[b];
    const int did = dom_ids[b];
    const float m1 = (att[b * SS + p1] == 1) ? 1.0f : 0.0f;
    const float m2 = (att[b * SS + p2] == 1) ? 1.0f : 0.0f;
    const float* r1 = emb + ((size_t)b * SS + p1) * DD;
    const float* r2 = emb + ((size_t)b * SS + p2) * DD;
    const float* dv = dset + (size_t)did * DD;
    for (int d = threadIdx.x; d < DD; d += blockDim.x) {
        const float dd = dv[d];
        pooled[b * K2 + d]      = r1[d] + m1 * dd;
        pooled[b * K2 + DD + d] = r2[d] + m2 * dd;
    }
}

// ------------------------------------------------- fused gate GEMM (WMMA) ---
// G[32, 6144] = pooled[32,1536] @ [w_ih_f; w_ih_b]^T + [b_f; b_b]
// One wave per 16-wide N-tile; each wave computes BOTH 16-row M-tiles so the
// bandwidth-dominant B (weight) fragment is loaded once and fed to 2 WMMAs.
__global__ __launch_bounds__(128)
void gate_gemm_wmma_kernel(const float* __restrict__ pooled,
                           const float* __restrict__ wf,
                           const float* __restrict__ bf,
                           const float* __restrict__ wb,
                           const float* __restrict__ bb,
                           float*       __restrict__ G)
{
    const int wave = threadIdx.x >> 5;
    const int lane = threadIdx.x & 31;
    const int nt   = blockIdx.x * 4 + wave;   // 0..383
    const int row  = lane & 15;               // M (for A) / N (for B) index
    const int hi   = lane >> 4;               // K sub-offset selector

    const float* W;  const float* bias;  int n0loc;
    if (nt < 192) { W = wf; bias = bf; n0loc = nt * 16; }
    else          { W = wb; bias = bb; n0loc = (nt - 192) * 16; }

    // A fragments: lane holds (M = m0+row, K = k + 2*hi .. +1) for m0 = 0, 16
    const float* aRow0 = pooled + (size_t)row        * K2 + 2 * hi;
    const float* aRow1 = pooled + (size_t)(16 + row) * K2 + 2 * hi;
    // B fragment: B[k][n] = W[n][k]; lane holds (N = n0+row, K = k + 2*hi .. +1)
    const float* bRow  = W + (size_t)(n0loc + row) * K2 + 2 * hi;

    v8f c0 = {};
    v8f c1 = {};
    #pragma unroll 8
    for (int k = 0; k < K2; k += 4) {
        v2f bv = *(const v2f*)(bRow + k);
        v2f a0 = *(const v2f*)(aRow0 + k);
        v2f a1 = *(const v2f*)(aRow1 + k);
        c0 = __builtin_amdgcn_wmma_f32_16x16x4_f32(
                false, a0, false, bv, (short)0, c0, false, false);
        c1 = __builtin_amdgcn_wmma_f32_16x16x4_f32(
                false, a1, false, bv, (short)0, c1, false, false);
    }

    // C/D layout: VGPR r -> M = r + 8*hi, N = row
    const int   jglob = nt * 16 + row;        // 0..6143 in concatenated space
    const float badd  = bias[n0loc + row];
    float* out0 = G + (size_t)(8 * hi)      * NCAT + jglob;   // M-tile 0
    float* out1 = G + (size_t)(16 + 8 * hi) * NCAT + jglob;   // M-tile 1
    #pragma unroll
    for (int r = 0; r < 8; ++r) {
        out0[(size_t)r * NCAT] = c0[r] + badd;
        out1[(size_t)r * NCAT] = c1[r] + badd;
    }
}

// ---------------------------------------------------------------- gates -----
// h[b,c] from gate pre-activations; split [i, f, g, o], f unused (c0 = 0).
__global__ void lstm_gates_kernel(const float* __restrict__ G,
                                  float*       __restrict__ h)
{
    const int idx = blockIdx.x * blockDim.x + threadIdx.x;
    if (idx >= NB * K2) return;
    const int b = idx / K2;
    const int c = idx % K2;
    const float* base = G + (size_t)b * NCAT + (c < DD ? 0 : G4);
    const int cc = (c < DD) ? c : c - DD;
    const float gi = base[cc];
    const float gg = base[2 * DD + cc];
    const float go = base[3 * DD + cc];
    const float si = 1.0f / (1.0f + expf(-gi));
    const float so = 1.0f / (1.0f + expf(-go));
    const float cn = si * tanhf(gg);
    h[idx] = so * tanhf(cn);
}

// ------------------------------------------------------------ head+argmax ---
__global__ __launch_bounds__(256)
void head_kernel(const float* __restrict__ h,
                 const float* __restrict__ wout,
                 const float* __restrict__ bout,
                 float*       __restrict__ out)
{
    __shared__ float red[NLAB][256];
    const int b = blockIdx.x, tid = threadIdx.x;
    float acc[NLAB];
    #pragma unroll
    for (int l = 0; l < NLAB; ++l) acc[l] = 0.0f;
    for (int k = tid; k < K2; k += 256) {
        const float hv = h[b * K2 + k];
        #pragma unroll
        for (int l = 0; l < NLAB; ++l)
            acc[l] += hv * wout[l * K2 + k];
    }
    #pragma unroll
    for (int l = 0; l < NLAB; ++l) red[l][tid] = acc[l];
    __syncthreads();
    if (tid < NLAB) {
        float s = 0.0f;
        for (int t = 0; t < 256; ++t) s += red[tid][t];
        s += bout[tid];
        out[b * NLAB + tid] = s;   // flat_logits
        red[tid][0] = s;
    }
    __syncthreads();
    if (tid == 0) {
        int best = 0; float bv = red[0][0];
        #pragma unroll
        for (int l = 1; l < NLAB; ++l)
            if (red[l][0] > bv) { bv = red[l][0]; best = l; }  // first-max wins
        out[NB * NLAB + b] = (float)best;   // labels appended after logits
    }
}

// ---------------------------------------------------------------- launch ----
extern "C" void kernel_launch(void* const* d_in, const int* in_sizes, int n_in,
                              void* d_out, int out_size, void* d_ws, size_t ws_size,
                              hipStream_t stream)
{
    const float* emb   = (const float*)d_in[0];   // [32,4096,768]
    const int*   att   = (const int*)  d_in[1];   // [32,4096]
    const int*   dids  = (const int*)  d_in[2];   // [32]
    const int*   e1    = (const int*)  d_in[3];   // [32]
    const int*   e2    = (const int*)  d_in[4];   // [32]
    const float* dset  = (const float*)d_in[5];   // [6,768]
    const float* wf    = (const float*)d_in[6];   // [3072,1536]
    const float* bf    = (const float*)d_in[7];   // [3072]
    const float* wb    = (const float*)d_in[8];   // [3072,1536]
    const float* bb    = (const float*)d_in[9];   // [3072]
    const float* wout  = (const float*)d_in[10];  // [17,1536]
    const float* bout  = (const float*)d_in[11];  // [17]
    float* out = (float*)d_out;                   // 544 logits + 32 labels

    float* pooled = (float*)d_ws;                 // [32,1536]
    float* G      = pooled + NB * K2;             // [32,6144]
    float* h      = G + NB * NCAT;                // [32,1536]

    gather_pooled_kernel<<<NB, 256, 0, stream>>>(emb, att, dids, e1, e2, dset, pooled);
    gate_gemm_wmma_kernel<<<96, 128, 0, stream>>>(pooled, wf, bf, wb, bb, G);
    lstm_gates_kernel<<<(NB * K2 + 255) / 256, 256, 0, stream>>>(G, h);
    head_kernel<<<NB, 256, 0, stream>>>(h, wout, bout, out);
}